// HDESN_37366215475888
// MI455X (gfx1250) — compile-verified
//
#include <hip/hip_runtime.h>

// ---------------- problem constants ----------------
#define T_STEPS 4096
#define IN_DIM  64
#define RES     2048
#define ROWS_PB 32                    // reservoir rows owned per block
#define NBLK    (RES / ROWS_PB)       // 64 persistent blocks (1 per WGP, 256KB LDS each)
#define TPB     256

typedef __attribute__((ext_vector_type(2))) float v2f;
typedef __attribute__((ext_vector_type(8))) float v8f;
typedef unsigned int u32x4 __attribute__((ext_vector_type(4)));
typedef int          i32x8 __attribute__((ext_vector_type(8)));
typedef int          i32x4 __attribute__((ext_vector_type(4)));

// TDM probe must only apply to the DEVICE pass; the x86 host pass of hipcc
// doesn't declare amdgcn builtins (that's what tripped the previous round).
#if defined(__HIP_DEVICE_COMPILE__)
#  if !defined(__has_builtin) || !__has_builtin(__builtin_amdgcn_tensor_load_to_lds)
#    error "gfx1250 DEVICE pass lacks __builtin_amdgcn_tensor_load_to_lds"
#  endif
#  define USE_TDM 1
#else
#  define USE_TDM 0   // host pass: parse the plain-copy variant instead
#endif

// ---------------------------------------------------------------------------
// Full-precision fp32 WMMA GEMM:  D(MxN) = A(MxK) * B(NxK)^T   (all row-major)
// One wave computes one 16x16 tile via V_WMMA_F32_16X16X4_F32, K-loop step 4.
// ---------------------------------------------------------------------------
__global__ void wmma_gemm_abT(const float* __restrict__ A,
                              const float* __restrict__ B,
                              float* __restrict__ D,
                              int M, int N, int K) {
    const int wid  = (int)((blockIdx.x * blockDim.x + threadIdx.x) >> 5);
    const int lane = (int)(threadIdx.x & 31);
    const int ntn  = N >> 4;
    const int tm   = wid / ntn;
    const int tn   = wid % ntn;
    if (tm * 16 >= M) return;

    const int half = lane >> 4;        // 0: K pair {0,1}, 1: K pair {2,3}
    const int idx  = lane & 15;
    const int koff = half * 2;

    const float* arow = A + (size_t)(tm * 16 + idx) * K;
    const float* brow = B + (size_t)(tn * 16 + idx) * K;

    v8f acc = {};
    for (int k = 0; k < K; k += 4) {
        v2f a, b;
        a.x = arow[k + koff];
        a.y = arow[k + koff + 1];
        b.x = brow[k + koff];
        b.y = brow[k + koff + 1];
        acc = __builtin_amdgcn_wmma_f32_16x16x4_f32(false, a, false, b,
                                                    (short)0, acc, false, false);
    }

    float* drow = D + (size_t)(tm * 16 + half * 8) * N + tn * 16 + idx;
#pragma unroll
    for (int v = 0; v < 8; ++v) drow[(size_t)v * N] = acc[v];
}

// ---------------------------------------------------------------------------
// y[t] = b + U[t,:] . dense_W[0:64]   (U part of feats = concat([U, H]))
// ---------------------------------------------------------------------------
__global__ void y_init(const float* __restrict__ U,
                       const float* __restrict__ w,
                       const float* __restrict__ b,
                       float* __restrict__ y) {
    int t = (int)(blockIdx.x * blockDim.x + threadIdx.x);
    if (t >= T_STEPS) return;
    float s = b[0];
    const float* u = U + (size_t)t * IN_DIM;
#pragma unroll
    for (int d = 0; d < IN_DIM; ++d) s = fmaf(u[d], w[d], s);
    y[t] = s;
}

// ---------------------------------------------------------------------------
// Zero h0 double-buffer + grid-barrier counter (every launch; ws is poisoned).
// ---------------------------------------------------------------------------
__global__ void ws_init(float* __restrict__ hbuf, unsigned int* __restrict__ bar) {
    int i = (int)(blockIdx.x * blockDim.x + threadIdx.x);
    if (i < 2 * RES) hbuf[i] = 0.0f;
    if (i == 0) *bar = 0u;
}

// ---------------------------------------------------------------------------
// Persistent recurrence. Each of the 64 blocks owns 32 rows of W_res, made
// LDS-resident (256KB) with ONE Tensor-Data-Mover descriptor (2D tile
// 32 x 2048 x 4B), then iterates the 4096 sequential steps.
// Per step, each wave processes 4 rows: lane l owns dwords l*4 + k*128 of a
// row (lanes 0-15 cover all 64 LDS banks; 2-way for the upper half = minimum
// for 512B/request), with the h vector hoisted into 64 VGPRs per wave.
// ---------------------------------------------------------------------------
__global__ void esn_recurrence(const float* __restrict__ Wres,
                               const float* __restrict__ Z,
                               const float* __restrict__ wh,   // dense_W + 64
                               float* __restrict__ hbuf,       // 2 x RES
                               float* __restrict__ y,
                               unsigned int* __restrict__ bar) {
    extern __shared__ float smem[];
    float* sW     = smem;                       // ROWS_PB * RES
    float* hstage = smem + ROWS_PB * RES;       // ROWS_PB
    float* swh    = hstage + ROWS_PB;           // ROWS_PB

    const int tid  = (int)threadIdx.x;
    const int lane = tid & 31;
    const int wv   = tid >> 5;                  // wave id 0..7
    const int r0   = (int)blockIdx.x * ROWS_PB;

#if USE_TDM
    // --- Tensor Data Mover: one 2D descriptor moves the whole 256KB slice ---
    if (wv == 0) {
        const unsigned long long ga =
            (unsigned long long)(const void*)(Wres + (size_t)r0 * RES);
        const unsigned lds_off = (unsigned)(unsigned long long)(void*)sW;
        u32x4 g0;
        g0[0] = 1u;                                              // count=1, load, user
        g0[1] = lds_off;                                         // lds_addr (bytes)
        g0[2] = (unsigned)(ga & 0xffffffffull);                  // global_addr[31:0]
        g0[3] = (unsigned)((ga >> 32) & 0x1ffffffull) | (2u << 30); // ga[56:32], type=2
        i32x8 g1;
        g1[0] = (int)(2u << 16);                 // data_size=2 -> 4 bytes/elem
        g1[1] = (int)(((unsigned)RES & 0xffffu) << 16);  // tensor_dim0[15:0]
        g1[2] = (int)(((unsigned)RES & 0xffffu) << 16);  // dim0[31:16]=0 | tensor_dim1[15:0]
        g1[3] = (int)((unsigned)RES << 16);      // dim1[31:16]=0 | tile_dim0=2048
        g1[4] = ROWS_PB;                         // tile_dim1=32, tile_dim2=0
        g1[5] = RES;                             // tensor_dim0_stride[31:0]
        g1[6] = 0;                               // stride[47:32]=0, dim1_stride lo=0
        g1[7] = 0;
        i32x4 gz = {0, 0, 0, 0};
#if __clang_major__ >= 23
        i32x8 gz8 = {0, 0, 0, 0, 0, 0, 0, 0};
        __builtin_amdgcn_tensor_load_to_lds(g0, g1, gz, gz, gz8, 0);
#else
        __builtin_amdgcn_tensor_load_to_lds(g0, g1, gz, gz, 0);
#endif
        __builtin_amdgcn_s_wait_tensorcnt(0);
    }
#else
    // host-pass parse-only fallback (device pass always uses the TDM path)
    for (int i = tid; i < ROWS_PB * (RES / 4); i += TPB) {
        const int row = i >> 9;                 // RES/4 = 512 float4 per row
        const int c4  = i & 511;
        float4 w4 = *(const float4*)(Wres + (size_t)(r0 + row) * RES + c4 * 4);
        *(float4*)(sW + (size_t)row * RES + c4 * 4) = w4;
    }
#endif
    if (tid < ROWS_PB) swh[tid] = wh[r0 + tid];
    __syncthreads();

    const float4* sW4 = (const float4*)sW;      // ROWS_PB * 512 float4s

    for (int t = 0; t < T_STEPS; ++t) {
        // hoist full h_{t-1} into registers (64 floats/lane, shared by 4 rows)
        const float4* hp4 = (const float4*)(hbuf + (size_t)(t & 1) * RES);
        float4 hreg[16];
#pragma unroll
        for (int k = 0; k < 16; ++k) hreg[k] = hp4[lane + k * 32];

        float hres[4];
#pragma unroll
        for (int j = 0; j < 4; ++j) {
            const int row = wv * 4 + j;
            float acc = 0.0f;
#pragma unroll
            for (int k = 0; k < 16; ++k) {
                float4 w4 = sW4[row * 512 + lane + k * 32];
                acc = fmaf(w4.x, hreg[k].x, acc);
                acc = fmaf(w4.y, hreg[k].y, acc);
                acc = fmaf(w4.z, hreg[k].z, acc);
                acc = fmaf(w4.w, hreg[k].w, acc);
            }
            acc += __shfl_down(acc, 16, 32);
            acc += __shfl_down(acc, 8, 32);
            acc += __shfl_down(acc, 4, 32);
            acc += __shfl_down(acc, 2, 32);
            acc += __shfl_down(acc, 1, 32);
            hres[j] = acc;                      // valid on lane 0
        }

        if (lane == 0) {
            float* hnext = hbuf + (size_t)((t + 1) & 1) * RES;
#pragma unroll
            for (int j = 0; j < 4; ++j) {
                const int row = wv * 4 + j;
                float h = tanhf(Z[(size_t)t * RES + r0 + row] + hres[j]);
                hnext[r0 + row] = h;
                hstage[row] = h;
            }
            if (t + 1 < T_STEPS)
                __builtin_prefetch(&Z[(size_t)(t + 1) * RES + r0 + wv * 4], 0, 0);
        }
        __syncthreads();

        // output-layer contribution for this block's 32 rows
        if (tid < ROWS_PB) {
            float v = hstage[tid] * swh[tid];
            v += __shfl_down(v, 16, 32);
            v += __shfl_down(v, 8, 32);
            v += __shfl_down(v, 4, 32);
            v += __shfl_down(v, 2, 32);
            v += __shfl_down(v, 1, 32);
            if (tid == 0) atomicAdd(&y[t], v);
        }

        // ---- device-wide step barrier (monotonic counter, no reset races) ----
        __threadfence();
        __syncthreads();
        if (tid == 0) {
            __hip_atomic_fetch_add(bar, 1u, __ATOMIC_ACQ_REL, __HIP_MEMORY_SCOPE_AGENT);
            const unsigned int target = (unsigned int)(t + 1) * (unsigned int)gridDim.x;
            while (__hip_atomic_load(bar, __ATOMIC_ACQUIRE, __HIP_MEMORY_SCOPE_AGENT) < target) {
                __builtin_amdgcn_s_sleep(8);
            }
            __threadfence();
        }
        __syncthreads();
    }
}

// ---------------------------------------------------------------------------
extern "C" void kernel_launch(void* const* d_in, const int* in_sizes, int n_in,
                              void* d_out, int out_size, void* d_ws, size_t ws_size,
                              hipStream_t stream) {
    (void)in_sizes; (void)n_in; (void)out_size; (void)ws_size;

    const float* X     = (const float*)d_in[0];   // (4096, 64, 1)
    const float* C     = (const float*)d_in[1];   // (64, 64)
    const float* W_in  = (const float*)d_in[2];   // (2048, 64)
    const float* W_res = (const float*)d_in[3];   // (2048, 2048)
    const float* dW    = (const float*)d_in[4];   // (1, 2112)
    const float* db    = (const float*)d_in[5];   // (1,)
    float* y = (float*)d_out;                     // (4096,)

    float* U    = (float*)d_ws;                          //  4096 x 64
    float* Z    = U + (size_t)T_STEPS * IN_DIM;          //  4096 x 2048
    float* hbuf = Z + (size_t)T_STEPS * RES;             //  2 x 2048
    unsigned int* bar = (unsigned int*)(hbuf + 2 * RES); //  1

    ws_init<<<dim3((2 * RES + 255) / 256), dim3(256), 0, stream>>>(hbuf, bar);

    // U = X * C^T   (M=4096, N=64, K=64) — fp32 WMMA
    {
        const int tiles = (T_STEPS / 16) * (IN_DIM / 16);
        wmma_gemm_abT<<<dim3(tiles * 32 / TPB), dim3(TPB), 0, stream>>>(
            X, C, U, T_STEPS, IN_DIM, IN_DIM);
    }
    // Z = U * W_in^T  (M=4096, N=2048, K=64) — fp32 WMMA
    {
        const int tiles = (T_STEPS / 16) * (RES / 16);
        wmma_gemm_abT<<<dim3(tiles * 32 / TPB), dim3(TPB), 0, stream>>>(
            U, W_in, Z, T_STEPS, RES, IN_DIM);
    }
    // y[t] = b + U[t] . dense_W[0:64]
    y_init<<<dim3(T_STEPS / 256), dim3(256), 0, stream>>>(U, dW, db, y);

    // sequential reservoir recurrence + output accumulation
    const size_t smem_bytes = (size_t)(ROWS_PB * RES + 2 * ROWS_PB) * sizeof(float);
    esn_recurrence<<<dim3(NBLK), dim3(TPB), smem_bytes, stream>>>(
        W_res, Z, dW + IN_DIM, hbuf, y, bar);
}